// GraphClassifier_85409719648935
// MI455X (gfx1250) — compile-verified
//
#include <hip/hip_runtime.h>

typedef __attribute__((ext_vector_type(2))) float v2f;
typedef __attribute__((ext_vector_type(8))) float v8f;

#define NNODES   50000
#define KNEI     20
#define NRELS    200          // padding row index (zeros) == 200
#define DIM      64
#define SEMD     32
#define EMB_STRIDE 65         // pad 64 -> 65 to break LDS bank conflicts
#define SEM_STRIDE 132        // pad 128 -> 132 (keeps 8B alignment for v2f)
#define WAVES    16
#define BLOCK    (WAVES * 32)

__device__ __forceinline__ float wave_reduce_add(float v) {
#pragma unroll
    for (int m = 16; m > 0; m >>= 1) v += __shfl_xor(v, m, 32);
    return v;
}

__device__ __forceinline__ float wave_reduce_max(float v) {
#pragma unroll
    for (int m = 16; m > 0; m >>= 1) v = fmaxf(v, __shfl_xor(v, m, 32));
    return v;
}

__global__ __launch_bounds__(BLOCK) void
rel_init_fused_kernel(const int* __restrict__ out_rels,
                      const int* __restrict__ in_rels,
                      const int* __restrict__ tgt_rels,
                      const float* __restrict__ feat,
                      const float* __restrict__ rel_emb,
                      const float* __restrict__ W,      // [32,128]
                      const float* __restrict__ bias,   // [32]
                      float* __restrict__ out)          // [N,64]
{
    __shared__ __align__(16) float ldsEmb[(NRELS + 1) * EMB_STRIDE]; // 52260 B
    __shared__ __align__(16) float ldsSem[WAVES * SEM_STRIDE];       //  8448 B

    const int tid  = threadIdx.x;
    const int lane = tid & 31;
    const int wave = tid >> 5;
    const int base = blockIdx.x * WAVES;
    const int node = base + wave;            // N is an exact multiple of 16

    // ---- Stage rel_emb (51 KB) into padded LDS once per block ----
    for (int i = tid; i < (NRELS + 1) * DIM; i += BLOCK)
        ldsEmb[(i >> 6) * EMB_STRIDE + (i & 63)] = rel_emb[i];
    __syncthreads();

    const int tbase = tgt_rels[node] * EMB_STRIDE;

    // ---- Per-node attention + aggregation, one wave per node ----
#pragma unroll
    for (int side = 0; side < 2; ++side) {
        const int* __restrict__ rels = side ? in_rels : out_rels;

        int r = NRELS;                                   // pad row (zeros)
        if (lane < KNEI) r = rels[node * KNEI + lane];
        const int rbase = r * EMB_STRIDE;

        // lane k computes score_k = <e_k, tgt> from LDS (conflict-free: stride 65)
        float s = 0.0f;
#pragma unroll 16
        for (int d = 0; d < DIM; ++d)
            s = __fmaf_rn(ldsEmb[rbase + d], ldsEmb[tbase + d], s);

        // softmax over lanes 0..19
        float mv  = wave_reduce_max((lane < KNEI) ? s : -3.0e38f);
        float e   = (lane < KNEI) ? __expf(s - mv) : 0.0f;
        float att = e / wave_reduce_add(e);

        // weighted aggregation: each lane owns dims (lane, lane+32)
        float acc0 = 0.0f, acc1 = 0.0f;
#pragma unroll
        for (int k = 0; k < KNEI; ++k) {
            const float a  = __shfl(att, k, 32);
            const int   rk = __shfl(r,   k, 32);
            acc0 = __fmaf_rn(a, ldsEmb[rk * EMB_STRIDE + lane],      acc0);
            acc1 = __fmaf_rn(a, ldsEmb[rk * EMB_STRIDE + 32 + lane], acc1);
        }
        ldsSem[wave * SEM_STRIDE + side * 64 + lane]      = acc0;
        ldsSem[wave * SEM_STRIDE + side * 64 + 32 + lane] = acc1;
    }

    // feat passthrough: out[n, 0..31] = feat[n, 0..31]
    out[node * 64 + lane] = feat[node * 32 + lane];

    __syncthreads();

    // ---- sem[16,128] @ W^T[128,32] via V_WMMA_F32_16X16X4_F32 ----
    // waves 0 and 1 each handle one 16-wide output tile (EXEC all-ones per wave)
    if (wave < 2) {
        const int ntile = wave;
        const int lo = lane & 15;
        const int hi = lane >> 4;
        const int col = ntile * 16 + lo;

        v8f c = {};
#pragma unroll
        for (int kk = 0; kk < 32; ++kk) {
            const int kb = kk * 4 + 2 * hi;  // A/B layout: VGPR v holds K = 4kk + 2*hi + v
            v2f a = *(const v2f*)&ldsSem[lo * SEM_STRIDE + kb];      // A[m=lo][kb..kb+1]
            v2f b = *(const v2f*)&W[col * 128 + kb];                 // B[k][n] = W[n][k]
            c = __builtin_amdgcn_wmma_f32_16x16x4_f32(
                    /*neg_a=*/false, a, /*neg_b=*/false, b,
                    /*c_mod=*/(short)0, c, /*reuse_a=*/false, /*reuse_b=*/false);
        }

        const float bc = bias[col];
#pragma unroll
        for (int rrow = 0; rrow < 8; ++rrow) {
            const int m = rrow + 8 * hi;     // C/D layout: VGPR r -> M = r + 8*hi
            float v = c[rrow] + bc;
            v = 1.0f / (1.0f + __expf(-v));  // sigmoid
            out[(base + m) * 64 + 32 + col] = v;
        }
    }
}

extern "C" void kernel_launch(void* const* d_in, const int* in_sizes, int n_in,
                              void* d_out, int out_size, void* d_ws, size_t ws_size,
                              hipStream_t stream) {
    (void)in_sizes; (void)n_in; (void)out_size; (void)d_ws; (void)ws_size;
    const int*   out_rels = (const int*)  d_in[0];
    const int*   in_rels  = (const int*)  d_in[1];
    const int*   tgt_rels = (const int*)  d_in[2];
    const float* feat     = (const float*)d_in[3];
    const float* rel_emb  = (const float*)d_in[4];
    const float* W        = (const float*)d_in[5];
    const float* bias     = (const float*)d_in[6];
    float*       out      = (float*)d_out;

    dim3 grid(NNODES / WAVES);  // 3125 blocks, 16 nodes each (exact)
    dim3 block(BLOCK);
    rel_init_fused_kernel<<<grid, block, 0, stream>>>(
        out_rels, in_rels, tgt_rels, feat, rel_emb, W, bias, out);
}